// GASLayer_86981677679238
// MI455X (gfx1250) — compile-verified
//
#include <hip/hip_runtime.h>
#include <cmath>

// Problem constants (from reference)
#define T_ROWS 32768
#define D_COLS 1024
#define KCHUNK 256      // number of chunks for the parallel scan
#define LROWS  128      // rows per chunk; KCHUNK*LROWS == T_ROWS

typedef float v2f __attribute__((ext_vector_type(2)));
typedef float v4f __attribute__((ext_vector_type(4)));
typedef float v8f __attribute__((ext_vector_type(8)));

// ---------------------------------------------------------------------------
// Pass A: per-chunk column reductions via WMMA f32 16x16x4.
//   For each chunk k and column d compute:
//     sum   = sum_i x[i][d]
//     sumsq = sum_i x[i][d]^2
//     W     = sum_i a^{-i} x[i][d]   (mu forcing; scaled by a^{L-1}*b later)
//   A-matrix trick: row0 = ones -> plain sum in C row 0
//                   row1 = a^{-(4j+k)} weights -> weighted sum in C row 1
//   One wave handles one chunk x 16 columns. EXEC is all-ones for WMMA.
// ---------------------------------------------------------------------------
__global__ void __launch_bounds__(256)
ema_pass_a(const float* __restrict__ x,
           float* __restrict__ sums, float* __restrict__ sumsqs,
           float* __restrict__ wsums,
           float ia, float ia2, float ia3, float ia4)
{
    const int lane = threadIdx.x & 31;
    const int wave = threadIdx.x >> 5;
    const int k    = blockIdx.x;
    const int col_base = (blockIdx.y * 8 + wave) * 16;
    const int t0   = k * LROWS;
    // B (4x16 f32) layout assumption (mirrors documented 16x4 A layout):
    //   VGPR0: lanes0-15 -> K=0, lanes16-31 -> K=2
    //   VGPR1: lanes0-15 -> K=1, lanes16-31 -> K=3
    const int row_v0 = (lane >> 4) * 2;          // 0 or 2
    const int col    = col_base + (lane & 15);

    // A (16x4 f32): lanes0-15 hold M=lane, K=0(v0)/1(v1); lanes16-31 K=2/3.
    const float one_m = ((lane & 15) == 0) ? 1.0f : 0.0f;   // row 0 = ones
    v2f a2; a2.x = one_m; a2.y = one_m;                      // for sumsq WMMA
    v2f a1;                                                  // row0 ones + row1 weights
    if (lane == 1)       { a1.x = 1.0f; a1.y = ia;  }
    else if (lane == 17) { a1.x = ia2;  a1.y = ia3; }
    else                 { a1.x = one_m; a1.y = one_m; }
    const float wscale = ((lane & 15) == 1) ? ia4 : 1.0f;    // advance a^{-4} per step

    v8f acc1 = {0.f,0.f,0.f,0.f,0.f,0.f,0.f,0.f};
    v8f acc2 = {0.f,0.f,0.f,0.f,0.f,0.f,0.f,0.f};

    const float* xp = x + (size_t)(t0 + row_v0) * D_COLS + col;
    #pragma unroll 4
    for (int j = 0; j < LROWS / 4; ++j) {
        __builtin_prefetch(xp + 16 * D_COLS, 0, 1);   // global_prefetch_b8
        const float b0 = xp[0];
        const float b1 = xp[D_COLS];
        v2f bx; bx.x = b0;      bx.y = b1;
        v2f bq; bq.x = b0 * b0; bq.y = b1 * b1;
        acc1 = __builtin_amdgcn_wmma_f32_16x16x4_f32(false, a1, false, bx,
                                                     (short)0, acc1, false, false);
        acc2 = __builtin_amdgcn_wmma_f32_16x16x4_f32(false, a2, false, bq,
                                                     (short)0, acc2, false, false);
        a1.x *= wscale; a1.y *= wscale;
        xp += 4 * D_COLS;
    }

    // C layout: VGPR0 lanes0-15 = row M=0 (N=lane); VGPR1 lanes0-15 = M=1.
    if (lane < 16) {
        const size_t o = (size_t)k * D_COLS + col;
        sums[o]   = acc1[0];
        wsums[o]  = acc1[1];
        sumsqs[o] = acc2[0];
    }
}

// ---------------------------------------------------------------------------
// Pass B: totals -> mu0, std0; then sequential combine of chunk mu-maps.
//   mu_start[k] = boundary mu entering chunk k.
// ---------------------------------------------------------------------------
__global__ void __launch_bounds__(256)
ema_pass_b(const float* __restrict__ sums, const float* __restrict__ sumsqs,
           const float* __restrict__ wsums,
           float* __restrict__ mu_start, float* __restrict__ std0_out,
           float aL, float bw)
{
    const int d = blockIdx.x * blockDim.x + threadIdx.x;
    if (d >= D_COLS) return;
    float ts = 0.f, tq = 0.f;
    for (int k = 0; k < KCHUNK; ++k) {
        ts += sums[(size_t)k * D_COLS + d];
        tq += sumsqs[(size_t)k * D_COLS + d];
    }
    const float mu0 = ts / (float)T_ROWS;
    float v = (tq - ts * mu0) / (float)(T_ROWS - 1);   // unbiased variance
    v = v < 0.f ? 0.f : v;
    std0_out[d] = sqrtf(v);                            // reference stores STD in var slot

    float mu = mu0;
    for (int k = 0; k < KCHUNK; ++k) {
        mu_start[(size_t)k * D_COLS + d] = mu;
        mu = aL * mu + bw * wsums[(size_t)k * D_COLS + d];
    }
}

// ---------------------------------------------------------------------------
// Pass C: replay mu inside each chunk, accumulate var forcing
//   F[k][d] = sum_i c^{L-1-i} (x_i - mu_i)^2   (recurrence f = c*f + dt^2)
// ---------------------------------------------------------------------------
__global__ void __launch_bounds__(256)
ema_pass_c(const float* __restrict__ x, const float* __restrict__ mu_start,
           float* __restrict__ F)
{
    const int k  = blockIdx.x;
    const int d4 = threadIdx.x * 4;
    const float b  = 0.01f;                 // ETA_MU
    const float cf = 1.0f - 0.02f;          // 1 - ETA_VAR
    const v4f* xr = reinterpret_cast<const v4f*>(x + (size_t)k * LROWS * D_COLS + d4);
    v4f mu = *reinterpret_cast<const v4f*>(mu_start + (size_t)k * D_COLS + d4);
    v4f f  = {0.f, 0.f, 0.f, 0.f};
    #pragma unroll 4
    for (int i = 0; i < LROWS; ++i) {
        __builtin_prefetch(xr + (i + 6) * (D_COLS / 4), 0, 1);
        v4f xv = xr[i * (D_COLS / 4)];
        mu = mu + b * (xv - mu);
        v4f dt = xv - mu;
        f = cf * f + dt * dt;
    }
    *reinterpret_cast<v4f*>(F + (size_t)k * D_COLS + d4) = f;
}

// ---------------------------------------------------------------------------
// Pass D: sequential combine of chunk var-maps.
// ---------------------------------------------------------------------------
__global__ void __launch_bounds__(256)
ema_pass_d(const float* __restrict__ F, const float* __restrict__ std0,
           float* __restrict__ var_start, float cL, float ev)
{
    const int d = blockIdx.x * blockDim.x + threadIdx.x;
    if (d >= D_COLS) return;
    float var = std0[d];
    for (int k = 0; k < KCHUNK; ++k) {
        var_start[(size_t)k * D_COLS + d] = var;
        var = cL * var + ev * F[(size_t)k * D_COLS + d];
    }
}

// ---------------------------------------------------------------------------
// Pass E: final replay per chunk writing norm, mu, var.
//   Non-temporal stores (402 MB streaming) so x stays resident in the 192MB L2.
// ---------------------------------------------------------------------------
__global__ void __launch_bounds__(256)
ema_pass_e(const float* __restrict__ x, const float* __restrict__ mu_start,
           const float* __restrict__ var_start,
           float* __restrict__ norm, float* __restrict__ info)
{
    const int k  = blockIdx.x;
    const int d4 = threadIdx.x * 4;
    const int t0 = k * LROWS;
    const float b  = 0.01f;                 // ETA_MU
    const float cf = 1.0f - 0.02f;          // 1 - ETA_VAR
    const float ev = 0.02f;                 // ETA_VAR
    const v4f* xr = reinterpret_cast<const v4f*>(x + (size_t)t0 * D_COLS + d4);
    v4f mu  = *reinterpret_cast<const v4f*>(mu_start  + (size_t)k * D_COLS + d4);
    v4f var = *reinterpret_cast<const v4f*>(var_start + (size_t)k * D_COLS + d4);
    #pragma unroll 2
    for (int i = 0; i < LROWS; ++i) {
        __builtin_prefetch(xr + (i + 6) * (D_COLS / 4), 0, 1);
        v4f xv = xr[i * (D_COLS / 4)];
        mu = mu + b * (xv - mu);
        v4f dt = xv - mu;
        var = cf * var + ev * (dt * dt);
        v4f nv;
        nv.x = dt.x / sqrtf(var.x);
        nv.y = dt.y / sqrtf(var.y);
        nv.z = dt.z / sqrtf(var.z);
        nv.w = dt.w / sqrtf(var.w);
        const size_t t = (size_t)(t0 + i);
        __builtin_nontemporal_store(nv,  reinterpret_cast<v4f*>(norm + t * D_COLS + d4));
        __builtin_nontemporal_store(mu,  reinterpret_cast<v4f*>(info + t * 2 * D_COLS + d4));
        __builtin_nontemporal_store(var, reinterpret_cast<v4f*>(info + t * 2 * D_COLS + D_COLS + d4));
    }
}

// ---------------------------------------------------------------------------
extern "C" void kernel_launch(void* const* d_in, const int* in_sizes, int n_in,
                              void* d_out, int out_size, void* d_ws, size_t ws_size,
                              hipStream_t stream) {
    (void)in_sizes; (void)n_in; (void)out_size; (void)ws_size;
    const float* x = (const float*)d_in[0];
    float* norm = (float*)d_out;                       // [T, D]
    float* info = (float*)d_out + (size_t)T_ROWS * D_COLS;  // [T, 2D]

    const size_t KD = (size_t)KCHUNK * D_COLS;
    float* ws       = (float*)d_ws;
    float* sums     = ws + 0 * KD;
    float* sumsqs   = ws + 1 * KD;
    float* wsums    = ws + 2 * KD;
    float* mu_start = ws + 3 * KD;
    float* F        = ws + 4 * KD;
    float* var_st   = ws + 5 * KD;
    float* std0     = ws + 6 * KD;   // [D]

    // Constants (double precision on host, then cast)
    const double aD  = (double)(1.0f - 0.01f);   // 0.99f
    const double cD  = (double)(1.0f - 0.02f);   // 0.98f
    const double iaD = 1.0 / aD;
    const float ia  = (float)iaD;
    const float ia2 = (float)(iaD * iaD);
    const float ia3 = (float)(iaD * iaD * iaD);
    const float ia4 = (float)(iaD * iaD * iaD * iaD);
    const float aL  = (float)pow(aD, (double)LROWS);
    const float bw  = (float)(0.01 * pow(aD, (double)(LROWS - 1)));  // b * a^{L-1}
    const float cL  = (float)pow(cD, (double)LROWS);

    ema_pass_a<<<dim3(KCHUNK, D_COLS / 128), 256, 0, stream>>>(
        x, sums, sumsqs, wsums, ia, ia2, ia3, ia4);
    ema_pass_b<<<dim3(D_COLS / 256), 256, 0, stream>>>(
        sums, sumsqs, wsums, mu_start, std0, aL, bw);
    ema_pass_c<<<dim3(KCHUNK), 256, 0, stream>>>(x, mu_start, F);
    ema_pass_d<<<dim3(D_COLS / 256), 256, 0, stream>>>(F, std0, var_st, cL, 0.02f);
    ema_pass_e<<<dim3(KCHUNK), 256, 0, stream>>>(x, mu_start, var_st, norm, info);
}